// UCALoraLinear_43258910605465
// MI455X (gfx1250) — compile-verified
//
#include <hip/hip_runtime.h>

typedef float v2f __attribute__((ext_vector_type(2)));
typedef float v8f __attribute__((ext_vector_type(8)));

#define NUM_CLASS 64
#define RANK      10
#define TOP_K     3
#define NCR       (NUM_CLASS * RANK)   // 640
#define BATCH     64
#define IN_F      1024
#define OUT_F     1024
#define SCALING   (16.0f / 10.0f)

// ---------------------------------------------------------------------------
// Kernel 1: per-batch top-3 mask, with class_mask (classes 0..6 zeroed) and
// the LoRA scaling folded in.  maskS[b, c*RANK + r] = SCALING or 0.
// ---------------------------------------------------------------------------
__global__ void uca_build_mask(const float* __restrict__ pseudo,
                               float* __restrict__ maskS) {
    const int b = threadIdx.x;
    if (b >= BATCH) return;
    const float* row = pseudo + b * NUM_CLASS;

    const float NEG = -3.402823466e38f;
    int s0 = -1, s1 = -1, s2 = -1;
    float bv = NEG;
    for (int c = 0; c < NUM_CLASS; ++c) { float t = row[c]; if (t > bv) { bv = t; s0 = c; } }
    bv = NEG;
    for (int c = 0; c < NUM_CLASS; ++c) { if (c == s0) continue; float t = row[c]; if (t > bv) { bv = t; s1 = c; } }
    bv = NEG;
    for (int c = 0; c < NUM_CLASS; ++c) { if (c == s0 || c == s1) continue; float t = row[c]; if (t > bv) { bv = t; s2 = c; } }

    float* mrow = maskS + b * NCR;
    for (int c = 0; c < NUM_CLASS; ++c) {
        const bool sel = (c == s0) | (c == s1) | (c == s2);
        const float val = (sel && c >= 7) ? SCALING : 0.0f;
        #pragma unroll
        for (int r = 0; r < RANK; ++r) mrow[c * RANK + r] = val;
    }
}

// ---------------------------------------------------------------------------
// 16x16 f32 wave tile (partial K) via V_WMMA_F32_16X16X4_F32.
// A slab: [16,K] row-major (lda); "B" slab: [16,K] row-major holding
// B[k][n] = Bt[n][k].  Four interleaved accumulators break the WMMA->WMMA
// RAW dependency chain.  K must be a multiple of 16.
// A-layout (ISA 7.12.2): lane<16 -> M=lane&15, K pair {0,1}; lane>=16 -> K {2,3}.
// B mirrors with N in the lane index.  D: VGPR j -> (M=j+8*(lane>>4), N=lane&15).
// ---------------------------------------------------------------------------
__device__ __forceinline__ v8f uca_gemm_tile_f32(const float* __restrict__ Ap,
                                                 const float* __restrict__ Bp,
                                                 int K, int lda, int ldb,
                                                 int lane) {
    const int lr   = lane & 15;
    const int half = lane >> 4;
    const float* arow = Ap + lr * lda + 2 * half;
    const float* brow = Bp + lr * ldb + 2 * half;

    v8f acc0 = {}; v8f acc1 = {}; v8f acc2 = {}; v8f acc3 = {};
    #pragma unroll 2
    for (int k = 0; k < K; k += 16) {
        v2f a0 = *(const v2f*)(arow + k);
        v2f b0 = *(const v2f*)(brow + k);
        v2f a1 = *(const v2f*)(arow + k + 4);
        v2f b1 = *(const v2f*)(brow + k + 4);
        v2f a2 = *(const v2f*)(arow + k + 8);
        v2f b2 = *(const v2f*)(brow + k + 8);
        v2f a3 = *(const v2f*)(arow + k + 12);
        v2f b3 = *(const v2f*)(brow + k + 12);
        acc0 = __builtin_amdgcn_wmma_f32_16x16x4_f32(false, a0, false, b0, (short)0, acc0, false, false);
        acc1 = __builtin_amdgcn_wmma_f32_16x16x4_f32(false, a1, false, b1, (short)0, acc1, false, false);
        acc2 = __builtin_amdgcn_wmma_f32_16x16x4_f32(false, a2, false, b2, (short)0, acc2, false, false);
        acc3 = __builtin_amdgcn_wmma_f32_16x16x4_f32(false, a3, false, b3, (short)0, acc3, false, false);
    }
    return acc0 + acc1 + acc2 + acc3;
}

// ---------------------------------------------------------------------------
// Kernel 2: tm[b,r] = (x @ lora_A^T)[b,r] * maskS[b,r]
// One 16x16 tile per block; 8 waves split K=1024 into 128-element chunks,
// reduce deterministically through LDS.  160 blocks -> 1280 waves.
// ---------------------------------------------------------------------------
__global__ void uca_tproj(const float* __restrict__ x,
                          const float* __restrict__ loraA,
                          const float* __restrict__ maskS,
                          float* __restrict__ tm) {
    __shared__ float part[8][256];
    const int wave = threadIdx.x >> 5;
    const int lane = threadIdx.x & 31;
    const int tn = blockIdx.x % (NCR / 16);
    const int tb = blockIdx.x / (NCR / 16);
    const int m0 = tb * 16, n0 = tn * 16;

    const int k0 = wave * (IN_F / 8);  // 128-wide K chunk per wave
    v8f acc = uca_gemm_tile_f32(x + m0 * IN_F + k0, loraA + n0 * IN_F + k0,
                                IN_F / 8, IN_F, IN_F, lane);
    #pragma unroll
    for (int j = 0; j < 8; ++j) part[wave][lane * 8 + j] = acc[j];
    __syncthreads();

    if (wave == 0) {
        v8f sum = {};
        #pragma unroll
        for (int w = 0; w < 8; ++w)
            #pragma unroll
            for (int j = 0; j < 8; ++j) sum[j] += part[w][lane * 8 + j];
        const int lr = lane & 15, half = lane >> 4;
        #pragma unroll
        for (int j = 0; j < 8; ++j) {
            const int m = m0 + j + 8 * half;
            const int n = n0 + lr;
            tm[m * NCR + n] = sum[j] * maskS[m * NCR + n];
        }
    }
}

// ---------------------------------------------------------------------------
// Kernel 3: out = x @ W_base^T + tm @ lora_B^T + 2*b_base
// One 16x16 tile per block (256 blocks).  Waves 0..3 split the main GEMM
// (K=1024, 256 each); waves 4..7 split the LoRA GEMM (K=640, 160 each).
// Deterministic LDS reduction.  256 blocks -> 2048 waves.
// ---------------------------------------------------------------------------
__global__ void uca_final(const float* __restrict__ x,
                          const float* __restrict__ Wb,
                          const float* __restrict__ bias,
                          const float* __restrict__ loraB,
                          const float* __restrict__ tm,
                          float* __restrict__ out) {
    __shared__ float part[8][256];
    const int wave = threadIdx.x >> 5;
    const int lane = threadIdx.x & 31;
    const int tn = blockIdx.x & (OUT_F / 16 - 1);
    const int tb = blockIdx.x / (OUT_F / 16);
    const int m0 = tb * 16, n0 = tn * 16;

    v8f acc;
    if (wave < 4) {
        const int k0 = wave * (IN_F / 4);      // 256-wide chunks of main K
        acc = uca_gemm_tile_f32(x + m0 * IN_F + k0, Wb + n0 * IN_F + k0,
                                IN_F / 4, IN_F, IN_F, lane);
    } else {
        const int k0 = (wave - 4) * (NCR / 4); // 160-wide chunks of LoRA K
        acc = uca_gemm_tile_f32(tm + m0 * NCR + k0, loraB + n0 * NCR + k0,
                                NCR / 4, NCR, NCR, lane);
    }
    #pragma unroll
    for (int j = 0; j < 8; ++j) part[wave][lane * 8 + j] = acc[j];
    __syncthreads();

    if (wave == 0) {
        v8f sum = {};
        #pragma unroll
        for (int w = 0; w < 8; ++w)
            #pragma unroll
            for (int j = 0; j < 8; ++j) sum[j] += part[w][lane * 8 + j];
        const int lr = lane & 15, half = lane >> 4;
        const float bb = 2.0f * bias[n0 + lr];   // bias added twice in the reference
        #pragma unroll
        for (int j = 0; j < 8; ++j) {
            const int m = m0 + j + 8 * half;
            out[m * OUT_F + (n0 + lr)] = sum[j] + bb;
        }
    }
}

// ---------------------------------------------------------------------------
extern "C" void kernel_launch(void* const* d_in, const int* in_sizes, int n_in,
                              void* d_out, int out_size, void* d_ws, size_t ws_size,
                              hipStream_t stream) {
    const float* x      = (const float*)d_in[0];
    const float* pseudo = (const float*)d_in[1];
    const float* Wb     = (const float*)d_in[2];
    const float* bias   = (const float*)d_in[3];
    const float* loraA  = (const float*)d_in[4];
    const float* loraB  = (const float*)d_in[5];
    float* out = (float*)d_out;

    float* maskS = (float*)d_ws;                       // 64*640 f32 = 160 KiB
    float* tm    = maskS + (size_t)BATCH * NCR;        // 64*640 f32 = 160 KiB

    uca_build_mask<<<1, 64, 0, stream>>>(pseudo, maskS);

    // one 16x16 tile per 256-thread block, 8-way split-K inside the block
    uca_tproj<<<(BATCH / 16) * (NCR / 16), 256, 0, stream>>>(x, loraA, maskS, tm);
    uca_final<<<(BATCH / 16) * (OUT_F / 16), 256, 0, stream>>>(x, Wb, bias, loraB, tm, out);
}